// SurgicalActionConditionedWorldModel_67989332296157
// MI455X (gfx1250) — compile-verified
//
#include <hip/hip_runtime.h>

typedef __attribute__((ext_vector_type(2))) float v2f;
typedef __attribute__((ext_vector_type(8))) float v8f;
typedef __attribute__((ext_vector_type(4))) unsigned int u32x4;
typedef __attribute__((ext_vector_type(4))) int i32x4;
typedef __attribute__((ext_vector_type(8))) int i32x8;

namespace {
constexpr int kT = 128;
constexpr int kB = 8;
constexpr int kH = 8;
constexpr int kD = 64;
constexpr int kPs = 68;   // LDS pitch (floats) for S state + Bm staging (pad 4 DW/row)
constexpr int kPa = 72;   // LDS pitch (floats) for A staging (pad 8 DW/row)
constexpr long kStep = (long)kB * kH * kD * kD;   // 262144 floats per t slice
}

#define HAVE_TDM (__has_builtin(__builtin_amdgcn_tensor_load_to_lds) && \
                  __has_builtin(__builtin_amdgcn_s_wait_tensorcnt))

// Per CDNA5 aperture mapping, the low 32 bits of a generic pointer into LDS
// are the LDS byte offset.
__device__ __forceinline__ unsigned lds_addr32(const void* p) {
  return (unsigned)(unsigned long long)p;
}

#if HAVE_TDM
// Issue one TDM DMA: 64x64 fp32 tile, row-major in memory (stride 64 floats),
// into LDS with pad_amount extra DWORDs appended after every 64-DWORD row.
// pad_code: 3 -> 4 DWORDs (pitch 68), 7 -> 8 DWORDs (pitch 72).
__device__ __forceinline__ void tdm_load_tile64(const float* gsrc,
                                                unsigned lds_byte,
                                                int pad_code) {
  const unsigned long long ga = (unsigned long long)gsrc;
  u32x4 g0;
  g0[0] = 1u;                                            // count=1, user D#
  g0[1] = lds_byte;                                      // lds_addr
  g0[2] = (unsigned)(ga & 0xFFFFFFFFull);                // global_addr[31:0]
  g0[3] = (unsigned)((ga >> 32) & 0x1FFFFFFull)          // global_addr[56:32]
          | (2u << 30);                                  // type = 2 (image)
  i32x8 g1;
  g1[0] = (2 << 16)            // data_size = 4 bytes
        | (1 << 20)            // pad_enable
        | (5 << 22)            // pad_interval: every 64 DWORDs
        | (pad_code << 25);    // pad_amount
  g1[1] = (64 << 16);          // tensor_dim0[15:0] = 64
  g1[2] = (64 << 16);          // tensor_dim1[15:0] = 64
  g1[3] = (64 << 16);          // tile_dim0 = 64
  g1[4] = 64;                  // tile_dim1 = 64 (tile_dim2 = 0)
  g1[5] = 64;                  // tensor_dim0_stride = 64 (low 32)
  g1[6] = 0;
  g1[7] = 0;
  i32x4 z4 = {0, 0, 0, 0};
#if __has_include(<hip/amd_detail/amd_gfx1250_TDM.h>)
  i32x8 z8 = {0, 0, 0, 0, 0, 0, 0, 0};
  __builtin_amdgcn_tensor_load_to_lds(g0, g1, z4, z4, z8, 0);   // 6-arg toolchain
#else
  __builtin_amdgcn_tensor_load_to_lds(g0, g1, z4, z4, 0);       // 5-arg toolchain
#endif
}
#else
// Fallback: cooperative synchronous staging with row padding (float4 chunks).
__device__ __forceinline__ void stage_tile64(const float* __restrict__ gsrc,
                                             float* __restrict__ dst,
                                             int pitch, int tid) {
  #pragma unroll
  for (int i = 0; i < 2; ++i) {
    const int q = tid * 2 + i;            // 1024 float4 chunks per tile
    const int row = q >> 4;
    const int c4  = (q & 15) * 4;
    const float4 v = *(const float4*)(gsrc + row * kD + c4);
    *(float4*)(dst + row * pitch + c4) = v;
  }
}
#endif

// One workgroup per (b,h) chain: 16 waves, each owning a 16x16 tile of the
// 64x64 state. S_t = S_{t-1} @ A_t + B_t via chained V_WMMA_F32_16X16X4_F32;
// A_t/Bm_t double-buffered in LDS by the Tensor Data Mover one step ahead.
__global__ __launch_bounds__(512)
void gated_delta_scan_kernel(const float* __restrict__ A,
                             const float* __restrict__ Bm,
                             const float* __restrict__ S0,
                             float* __restrict__ out) {
  __shared__ __align__(16) float sS[2][kD * kPs];   // state ping-pong
  __shared__ __align__(16) float sA[2][kD * kPa];   // staged A_t
  __shared__ __align__(16) float sB[2][kD * kPs];   // staged Bm_t

  const int tid  = threadIdx.x;
  const int lane = tid & 31;
  const int wv   = tid >> 5;       // wave id 0..15
  const int mi   = wv >> 2;        // tile row of S
  const int ni   = wv & 3;         // tile col of S
  const int l15  = lane & 15;
  const int lh   = lane >> 4;      // lane half

  const int  bh        = blockIdx.x;               // chain id 0..63
  const long chainBase = (long)bh * (kD * kD);
  const float* Abase = A  + chainBase;
  const float* Bbase = Bm + chainBase;

  // ---- stage initial state into sS[0] (coalesced) ----
  const float* s0p = S0 + chainBase;
  #pragma unroll
  for (int j = 0; j < 8; ++j) {
    const int idx = tid * 8 + j;
    sS[0][(idx >> 6) * kPs + (idx & 63)] = s0p[idx];
  }

  // ---- stage A_0 / Bm_0 into buffer 0 ----
#if HAVE_TDM
  if (wv == 0) {
    tdm_load_tile64(Abase, lds_addr32(&sA[0][0]), 7);
    tdm_load_tile64(Bbase, lds_addr32(&sB[0][0]), 3);
    __builtin_amdgcn_s_wait_tensorcnt(0);
  }
#else
  stage_tile64(Abase, sA[0], kPa, tid);
  stage_tile64(Bbase, sB[0], kPs, tid);
#endif
  __syncthreads();

  // C/D layout coordinates: VGPR v holds (row = crow + v, col = ccol)
  const int crow = 16 * mi + 8 * lh;
  const int ccol = 16 * ni + l15;

  for (int t = 0; t < kT; ++t) {
    const int cur = t & 1;
    const int nxt = cur ^ 1;

    // Kick off DMA of step t+1 operands; overlaps with the compute below.
    if (t + 1 < kT) {
#if HAVE_TDM
      if (wv == 0) {
        tdm_load_tile64(Abase + (long)(t + 1) * kStep, lds_addr32(&sA[nxt][0]), 7);
        tdm_load_tile64(Bbase + (long)(t + 1) * kStep, lds_addr32(&sB[nxt][0]), 3);
      }
#else
      stage_tile64(Abase + (long)(t + 1) * kStep, sA[nxt], kPa, tid);
      stage_tile64(Bbase + (long)(t + 1) * kStep, sB[nxt], kPs, tid);
#endif
    }

    // Seed accumulator with the additive-update tile (C layout, from LDS).
    v8f c;
    #pragma unroll
    for (int v = 0; v < 8; ++v)
      c[v] = sB[cur][(crow + v) * kPs + ccol];

    // 16 chained K=4 fp32 WMMAs: rows of S_{t-1} (A-op) x cols of A_t (B-op).
    const float* sp  = &sS[cur][(16 * mi + l15) * kPs];
    const float* apB = &sA[cur][0];
    #pragma unroll
    for (int kk = 0; kk < 16; ++kk) {
      v2f aop, bop;
      aop.x = sp[4 * kk + 2 * lh + 0];
      aop.y = sp[4 * kk + 2 * lh + 1];
      bop.x = apB[(4 * kk + 2 * lh + 0) * kPa + ccol];
      bop.y = apB[(4 * kk + 2 * lh + 1) * kPa + ccol];
      c = __builtin_amdgcn_wmma_f32_16x16x4_f32(
              /*neg_a=*/false, aop, /*neg_b=*/false, bop,
              /*c_mod=*/(short)0, c, /*reuse_a=*/false, /*reuse_b=*/false);
    }

    // Publish S_t: LDS ping-pong for the next step + global output tile.
    float* Ot = out + chainBase + (long)t * kStep;
    #pragma unroll
    for (int v = 0; v < 8; ++v) {
      sS[nxt][(crow + v) * kPs + ccol] = c[v];
      Ot[(crow + v) * kD + ccol] = c[v];
    }

#if HAVE_TDM
    if (wv == 0) __builtin_amdgcn_s_wait_tensorcnt(0);  // t+1 tiles landed
#endif
    __syncthreads();
  }
}

extern "C" void kernel_launch(void* const* d_in, const int* in_sizes, int n_in,
                              void* d_out, int out_size, void* d_ws, size_t ws_size,
                              hipStream_t stream) {
  const float* A   = (const float*)d_in[0];   // [T,B,H,D,D]
  const float* Bm  = (const float*)d_in[1];   // [T,B,H,D,D]
  const float* S0  = (const float*)d_in[2];   // [B,H,D,D]
  float*       out = (float*)d_out;           // [T,B,H,D,D]

  (void)in_sizes; (void)n_in; (void)out_size; (void)d_ws; (void)ws_size;

  dim3 grid(kB * kH);   // 64 independent chains
  dim3 block(512);      // 16 waves (wave32)
  hipLaunchKernelGGL(gated_delta_scan_kernel, grid, block, 0, stream, A, Bm, S0, out);
}